// GAT_10892037062910
// MI455X (gfx1250) — compile-verified
//
#include <hip/hip_runtime.h>
#include <math.h>

// ---------------- problem constants (from reference) ----------------
#define IN_C      128
#define HID       8
#define HEADS     8
#define OUT_C     16
#define NEG_SLOPE 0.2f
#define EPS_      1e-16f

typedef __attribute__((ext_vector_type(2))) float v2f;
typedef __attribute__((ext_vector_type(8))) float v8f;

// ---------------- utility ----------------
__global__ void fill_kernel(float* __restrict__ p, long long n, float v) {
    long long i = (long long)blockIdx.x * blockDim.x + threadIdx.x;
    if (i < n) p[i] = v;
}

// ---------------- dense transform: Out[M,NC] = A[M,K] @ W[K,NC] ----------------
// One block handles 16 rows; wave w handles column tile [16w, 16w+16).
// V_WMMA_F32_16X16X4_F32 (full-fp32 matrix pipe, exact dtype match with ref).
// fp32 WMMA layouts (ISA 7.12.2):
//   A 16x4 : lanes 0-15 -> M=lane, K={0,1} in v0/v1 ; lanes 16-31 -> M=lane-16, K={2,3}
//   B 4x16 : lanes 0-15 -> N=lane, K={0,1} in v0/v1 ; lanes 16-31 -> N=lane-16, K={2,3}
//   D 16x16: VGPR r -> (M=r, N=lane) lanes 0-15 ; (M=r+8, N=lane-16) lanes 16-31
//
// W is stored TRANSPOSED in LDS (Ws[col][k]) so both A and B fragments are
// consecutive float pairs -> single 8-byte LDS loads with immediate offsets.
// Rows padded by 4 floats: stride K+4 breaks the K%64==0 bank-conflict pattern
// (bank = (4n + k) % 64, distinct for the 16 lanes) and stays 8B-aligned.
template <int K, int NC>
__global__ __launch_bounds__(32 * (NC / 16))
void gemm_wmma_kernel(const float* __restrict__ A,
                      const float* __restrict__ W,
                      float* __restrict__ Out,
                      int M) {
    constexpr int KP = K + 4;
    __shared__ float As[16 * KP];   // A tile, padded rows
    __shared__ float Ws[NC * KP];   // W transposed, padded rows

    const int row0 = blockIdx.x * 16;

    // cooperative stage: A tile (coalesced read, padded LDS write)
    for (int i = threadIdx.x; i < 16 * K; i += blockDim.x) {
        int r = i / K, c = i % K;
        int row = row0 + r;
        As[r * KP + c] = (row < M) ? A[(size_t)row * K + c] : 0.0f;
    }
    // cooperative stage: W (coalesced global read, transposed LDS write)
    for (int i = threadIdx.x; i < K * NC; i += blockDim.x) {
        int k = i / NC, c = i % NC;
        Ws[c * KP + k] = W[i];
    }
    __syncthreads();

    const int wv   = threadIdx.x >> 5;   // column tile index
    const int lane = threadIdx.x & 31;
    const int n    = lane & 15;          // N (and M for A frag) within tile
    const int kh   = lane >> 4;          // lane-half -> K offset 0 or 2
    const int col  = wv * 16 + n;

    const float* aRow = As + n * KP + 2 * kh;     // per-lane base, then imm offsets
    const float* bRow = Ws + col * KP + 2 * kh;

    v8f acc = {};
#pragma unroll
    for (int k4 = 0; k4 < K; k4 += 4) {
        v2f a = *(const v2f*)(aRow + k4);
        v2f b = *(const v2f*)(bRow + k4);
        acc = __builtin_amdgcn_wmma_f32_16x16x4_f32(
            /*neg_a=*/false, a, /*neg_b=*/false, b,
            /*c_mod=*/(short)0, acc, /*reuse_a=*/false, /*reuse_b=*/false);
    }

#pragma unroll
    for (int r = 0; r < 8; ++r) {
        int row = row0 + r + 8 * kh;
        if (row < M) Out[(size_t)row * NC + col] = acc[r];
    }
}

// ---------------- per-(node,head) attention coefficients ----------------
// a_src[n,h] = sum_c h[n,h,c]*att_src[h,c] ; same for dst.
__global__ void attn_coef_kernel(const float* __restrict__ h,
                                 const float* __restrict__ att_src,
                                 const float* __restrict__ att_dst,
                                 float* __restrict__ a_s,
                                 float* __restrict__ a_d,
                                 int Nn, int H, int C) {
    long long i = (long long)blockIdx.x * blockDim.x + threadIdx.x;
    if (i >= (long long)Nn * H) return;
    int node = (int)(i / H), hd = (int)(i % H);
    const float* hp = h + (size_t)node * H * C + (size_t)hd * C;
    const float* as = att_src + hd * C;
    const float* ad = att_dst + hd * C;
    float s = 0.0f, d = 0.0f;
    for (int c = 0; c < C; ++c) {
        float v = hp[c];
        s += v * as[c];
        d += v * ad[c];
    }
    a_s[i] = s;
    a_d[i] = d;
}

__device__ __forceinline__ void edge_sd(const long long* __restrict__ src,
                                        const long long* __restrict__ dst,
                                        long long e, long long E,
                                        long long& s, long long& d) {
    if (e < E) { s = src[e]; d = dst[e]; }
    else       { s = d = e - E; }           // self-loops appended after E edges
}

// ---------------- pass 1: segment max of leaky-relu'd logits ----------------
__global__ void edge_max_kernel(const long long* __restrict__ src,
                                const long long* __restrict__ dst,
                                long long E, long long Etot,
                                const float* __restrict__ a_s,
                                const float* __restrict__ a_d,
                                float* __restrict__ amax, int H) {
    long long i = (long long)blockIdx.x * blockDim.x + threadIdx.x;
    if (i >= Etot * H) return;
    long long e = i / H;
    int hd = (int)(i % H);
    long long s, d;
    edge_sd(src, dst, e, E, s, d);
    float al = a_s[s * H + hd] + a_d[d * H + hd];
    al = (al > 0.0f) ? al : NEG_SLOPE * al;
    atomicMax(&amax[d * H + hd], al);      // global_atomic_max_num_f32
}

// ---------------- pass 2: un-normalized softmax weight + weighted scatter ----------------
// Division by asum is per-(dst,head), so it is deferred to finalize.
__global__ void edge_accum_kernel(const long long* __restrict__ src,
                                  const long long* __restrict__ dst,
                                  long long E, long long Etot,
                                  const float* __restrict__ a_s,
                                  const float* __restrict__ a_d,
                                  const float* __restrict__ amax,
                                  float* __restrict__ asum,
                                  const float* __restrict__ hfeat,
                                  float* __restrict__ agg,
                                  int H, int C) {
    long long i = (long long)blockIdx.x * blockDim.x + threadIdx.x;
    if (i >= Etot * H) return;
    long long e = i / H;
    int hd = (int)(i % H);
    long long s, d;
    edge_sd(src, dst, e, E, s, d);
    float al = a_s[s * H + hd] + a_d[d * H + hd];
    al = (al > 0.0f) ? al : NEG_SLOPE * al;
    float w = expf(al - amax[d * H + hd]);
    atomicAdd(&asum[d * H + hd], w);
    const float* hs = hfeat + (size_t)s * H * C + (size_t)hd * C;
    float* od = agg + (size_t)d * H * C + (size_t)hd * C;
    for (int c = 0; c < C; ++c) atomicAdd(&od[c], w * hs[c]);
}

// ---------------- finalize: /(asum+eps) + bias, relu ----------------
__global__ void finalize_kernel(const float* __restrict__ agg,
                                const float* __restrict__ asum,
                                const float* __restrict__ bias,
                                float* __restrict__ out,
                                int Nn, int H, int C) {
    long long i = (long long)blockIdx.x * blockDim.x + threadIdx.x;
    long long tot = (long long)Nn * H * C;
    if (i >= tot) return;
    int node = (int)(i / (H * C));
    int f    = (int)(i % (H * C));
    int hd   = f / C;
    float v = agg[i] / (asum[(size_t)node * H + hd] + EPS_) + bias[f];
    out[i] = (v > 0.0f) ? v : 0.0f;
}

// ---------------- host-side orchestration ----------------
extern "C" void kernel_launch(void* const* d_in, const int* in_sizes, int n_in,
                              void* d_out, int out_size, void* d_ws, size_t ws_size,
                              hipStream_t stream) {
    const float*     x        = (const float*)d_in[0];
    const long long* ei       = (const long long*)d_in[1];
    const float*     W1       = (const float*)d_in[2];
    const float*     att_src1 = (const float*)d_in[3];
    const float*     att_dst1 = (const float*)d_in[4];
    const float*     b1       = (const float*)d_in[5];
    const float*     W2       = (const float*)d_in[6];
    const float*     att_src2 = (const float*)d_in[7];
    const float*     att_dst2 = (const float*)d_in[8];
    const float*     b2       = (const float*)d_in[9];

    const int       Nn   = in_sizes[0] / IN_C;        // 100000 (multiple of 16)
    const long long E    = (long long)in_sizes[1] / 2;
    const long long Etot = E + Nn;
    const long long* srcI = ei;         // edge_index[0]
    const long long* dstI = ei + E;     // edge_index[1]

    // Workspace layout (floats). Layer-2 buffers overlay the dead h1 region.
    float* ws    = (float*)d_ws;
    float* h1    = ws;                              // N*64
    float* a_s1  = h1    + (size_t)Nn * 64;         // N*8
    float* a_d1  = a_s1  + (size_t)Nn * 8;          // N*8
    float* amax1 = a_d1  + (size_t)Nn * 8;          // N*8
    float* asum1 = amax1 + (size_t)Nn * 8;          // N*8
    float* agg1  = asum1 + (size_t)Nn * 8;          // N*64 (becomes y1 in-place)
    // layer-2 overlays (N*36 <= N*64)
    float* h2    = h1;                              // N*16
    float* a_s2  = h2    + (size_t)Nn * 16;         // N
    float* a_d2  = a_s2  + Nn;                      // N
    float* amax2 = a_d2  + Nn;                      // N
    float* asum2 = amax2 + Nn;                      // N
    float* agg2  = asum2 + Nn;                      // N*16

    const int TB = 256;
    auto cdiv = [](long long a, long long b) -> int { return (int)((a + b - 1) / b); };

    // ================= layer 1 =================
    fill_kernel<<<cdiv((long long)Nn * 8,  TB), TB, 0, stream>>>(amax1, (long long)Nn * 8,  -INFINITY);
    fill_kernel<<<cdiv((long long)Nn * 8,  TB), TB, 0, stream>>>(asum1, (long long)Nn * 8,  0.0f);
    fill_kernel<<<cdiv((long long)Nn * 64, TB), TB, 0, stream>>>(agg1,  (long long)Nn * 64, 0.0f);

    gemm_wmma_kernel<IN_C, HEADS * HID>
        <<<(Nn + 15) / 16, 32 * ((HEADS * HID) / 16), 0, stream>>>(x, W1, h1, Nn);

    attn_coef_kernel<<<cdiv((long long)Nn * HEADS, TB), TB, 0, stream>>>(
        h1, att_src1, att_dst1, a_s1, a_d1, Nn, HEADS, HID);
    edge_max_kernel<<<cdiv(Etot * HEADS, TB), TB, 0, stream>>>(
        srcI, dstI, E, Etot, a_s1, a_d1, amax1, HEADS);
    edge_accum_kernel<<<cdiv(Etot * HEADS, TB), TB, 0, stream>>>(
        srcI, dstI, E, Etot, a_s1, a_d1, amax1, asum1, h1, agg1, HEADS, HID);
    finalize_kernel<<<cdiv((long long)Nn * 64, TB), TB, 0, stream>>>(
        agg1, asum1, b1, agg1, Nn, HEADS, HID);                  // y1 = relu(agg/asum + b1)

    // ================= layer 2 =================
    fill_kernel<<<cdiv((long long)Nn,      TB), TB, 0, stream>>>(amax2, (long long)Nn,      -INFINITY);
    fill_kernel<<<cdiv((long long)Nn,      TB), TB, 0, stream>>>(asum2, (long long)Nn,      0.0f);
    fill_kernel<<<cdiv((long long)Nn * 16, TB), TB, 0, stream>>>(agg2,  (long long)Nn * 16, 0.0f);

    gemm_wmma_kernel<HEADS * HID, OUT_C>
        <<<(Nn + 15) / 16, 32 * (OUT_C / 16), 0, stream>>>(agg1, W2, h2, Nn);

    attn_coef_kernel<<<cdiv((long long)Nn, TB), TB, 0, stream>>>(
        h2, att_src2, att_dst2, a_s2, a_d2, Nn, 1, OUT_C);
    edge_max_kernel<<<cdiv(Etot, TB), TB, 0, stream>>>(
        srcI, dstI, E, Etot, a_s2, a_d2, amax2, 1);
    edge_accum_kernel<<<cdiv(Etot, TB), TB, 0, stream>>>(
        srcI, dstI, E, Etot, a_s2, a_d2, amax2, asum2, h2, agg2, 1, OUT_C);
    finalize_kernel<<<cdiv((long long)Nn * OUT_C, TB), TB, 0, stream>>>(
        agg2, asum2, b2, (float*)d_out, Nn, 1, OUT_C);
}